// Net_30897994728156
// MI455X (gfx1250) — compile-verified
//
#include <hip/hip_runtime.h>
#include <hip/hip_bf16.h>
#include <math.h>

// ---- problem dims (fixed by reference) ----
#define Nn   100000
#define Ee   3200000
#define C1n  25000
#define E2n  800000
#define C2n  6250
#define Bn   64
// FIN=3, H1=16, H2=32, HFC=64

typedef __attribute__((ext_vector_type(2))) float v2f;
typedef __attribute__((ext_vector_type(8))) float v8f;

// -------------------- utility --------------------

__global__ void fill_kernel(float* __restrict__ p, float v, long n) {
    long i = (long)blockIdx.x * blockDim.x + threadIdx.x;
    if (i < n) p[i] = v;
}

__device__ __forceinline__ void atomicMaxFloat(float* addr, float val) {
    // classic signed/unsigned trick; correct for any float ordering, init to -inf
    if (val >= 0.0f) {
        atomicMax((int*)addr, __float_as_int(val));
    } else {
        atomicMin((unsigned int*)addr, __float_as_uint(val));
    }
}

// -------------------- WMMA GEMMs --------------------
// A layout (32-bit A 16x4): lane L: M = L%16; VGPR0/1 hold K = (L<16 ? 0,1 : 2,3).
// B layout (32-bit B 4x16): lane L: N = L%16; VGPR0/1 hold K = (L<16 ? 0,1 : 2,3).
// C/D layout: VGPR r: lanes 0-15 -> M=r, N=lane; lanes 16-31 -> M=r+8, N=lane-16.

// h[N,16] = x[N,3] @ W1[3,16]; N is an exact multiple of 16 -> branch-free.
// K=3 padded to 4 with a 0/1 mask on clamped (in-range) indices.
__global__ __launch_bounds__(32) void gemm_x_w1(const float* __restrict__ x,
                                                const float* __restrict__ W1,
                                                float* __restrict__ h) {
    const int lane = threadIdx.x;
    const int tile = blockIdx.x;
    const int col  = lane & 15;
    const int row  = tile * 16 + (lane & 15);
    const int k0   = (lane >> 4) * 2;            // 0 or 2
    const int k1   = (lane < 16) ? 1 : 2;        // clamped in-range index for K=k0+1
    const float mk = (lane < 16) ? 1.0f : 0.0f;  // zero out padded K=3 lane half

    v2f a, b;
    a.x = x[row * 3 + k0];
    a.y = x[row * 3 + k1] * mk;
    b.x = W1[k0 * 16 + col];
    b.y = W1[k1 * 16 + col] * mk;

    v8f c = {};
    c = __builtin_amdgcn_wmma_f32_16x16x4_f32(false, a, false, b, (short)0, c, false, false);

    const int mbase = tile * 16 + ((lane < 16) ? 0 : 8);
#pragma unroll
    for (int r = 0; r < 8; ++r) {
        h[(mbase + r) * 16 + col] = c[r];
    }
}

// h2[C1,32] = hp[C1,16] @ W2[16,32]: 4 K-steps x 2 column tiles (A reused) = 8 WMMAs.
// Deliberately unguarded: padded reads/writes for the last partial row tile stay
// inside the workspace (see layout in kernel_launch); junk rows are never consumed.
__global__ __launch_bounds__(32) void gemm_hp_w2(const float* __restrict__ hp,
                                                 const float* __restrict__ W2,
                                                 float* __restrict__ h2) {
    const int lane = threadIdx.x;
    const int tile = blockIdx.x;      // row tile
    const int col  = lane & 15;
    const int row  = tile * 16 + (lane & 15);
    const int kh   = (lane >> 4) * 2; // 0 or 2 within each K-chunk of 4

    v8f c0 = {};
    v8f c1 = {};
#pragma unroll
    for (int kc = 0; kc < 16; kc += 4) {
        v2f a, b0, b1;
        a.x  = hp[row * 16 + kc + kh + 0];
        a.y  = hp[row * 16 + kc + kh + 1];
        b0.x = W2[(kc + kh + 0) * 32 + col];
        b0.y = W2[(kc + kh + 1) * 32 + col];
        b1.x = W2[(kc + kh + 0) * 32 + 16 + col];
        b1.y = W2[(kc + kh + 1) * 32 + 16 + col];
        c0 = __builtin_amdgcn_wmma_f32_16x16x4_f32(false, a, false, b0, (short)0, c0, false, false);
        c1 = __builtin_amdgcn_wmma_f32_16x16x4_f32(false, a, false, b1, (short)0, c1, false, false);
    }

    const int mbase = tile * 16 + ((lane < 16) ? 0 : 8);
#pragma unroll
    for (int r = 0; r < 8; ++r) {
        h2[(mbase + r) * 32 + col]      = c0[r];
        h2[(mbase + r) * 32 + 16 + col] = c1[r];
    }
}

// -------------------- edge kernels (bandwidth-bound) --------------------

template <int D>
__global__ void edge_logits_max(const float* __restrict__ h,
                                const int* __restrict__ ei,
                                const float* __restrict__ att,
                                float* __restrict__ logits,
                                float* __restrict__ m, int E) {
    int e = blockIdx.x * blockDim.x + threadIdx.x;
    if (e >= E) return;
    int src = ei[e];
    int dst = ei[E + e];
    const float4* hd = (const float4*)(h + (long)dst * D);
    const float4* hs = (const float4*)(h + (long)src * D);
    const float4* wd = (const float4*)att;        // att[:D]   (dst part)
    const float4* ws = (const float4*)(att + D);  // att[D:2D] (src part)
    float l = 0.0f;
#pragma unroll
    for (int q = 0; q < D / 4; ++q) {
        float4 a = hd[q], b = hs[q], wa = wd[q], wb = ws[q];
        l += a.x * wa.x + a.y * wa.y + a.z * wa.z + a.w * wa.w;
        l += b.x * wb.x + b.y * wb.y + b.z * wb.z + b.w * wb.w;
    }
    l = (l > 0.0f) ? l : 0.2f * l;   // leaky_relu(0.2)
    logits[e] = l;
    atomicMaxFloat(&m[dst], l);
}

__global__ void edge_exp_sum(const int* __restrict__ ei,
                             float* __restrict__ logits,
                             const float* __restrict__ m,
                             float* __restrict__ s, int E) {
    int e = blockIdx.x * blockDim.x + threadIdx.x;
    if (e >= E) return;
    int dst = ei[E + e];
    float ex = expf(logits[e] - m[dst]);  // m[dst] finite: dst has >=1 edge
    logits[e] = ex;
    atomicAdd(&s[dst], ex);
}

template <int D>
__global__ void edge_aggregate(const float* __restrict__ h,
                               const int* __restrict__ ei,
                               const float* __restrict__ w_e,
                               const float* __restrict__ logits,
                               const float* __restrict__ s,
                               float* __restrict__ out, int E) {
    int e = blockIdx.x * blockDim.x + threadIdx.x;
    if (e >= E) return;
    int src = ei[e];
    int dst = ei[E + e];
    float alpha = (logits[e] / s[dst]) * w_e[e];
    const float4* hs = (const float4*)(h + (long)src * D);
    float* od = out + (long)dst * D;
#pragma unroll
    for (int q = 0; q < D / 4; ++q) {
        float4 v = hs[q];
        atomicAdd(&od[q * 4 + 0], alpha * v.x);
        atomicAdd(&od[q * 4 + 1], alpha * v.y);
        atomicAdd(&od[q * 4 + 2], alpha * v.z);
        atomicAdd(&od[q * 4 + 3], alpha * v.w);
    }
}

// -------------------- pooling / readout --------------------

// v = relu(out + b); pooled[cluster] = max(pooled, v). v >= 0 so uint-bit atomicMax
// is exact with pooled initialized to 0 (matches segment_max-with-0-for-empty).
template <int D>
__global__ void bias_relu_pool(const float* __restrict__ out,
                               const float* __restrict__ b,
                               const int* __restrict__ cluster,
                               float* __restrict__ pooled, int n) {
    long idx = (long)blockIdx.x * blockDim.x + threadIdx.x;
    if (idx >= (long)n * D) return;
    int node = (int)(idx / D);
    int k    = (int)(idx % D);
    float v  = out[idx] + b[k];
    v = (v > 0.0f) ? v : 0.0f;
    atomicMax((unsigned int*)&pooled[(long)cluster[node] * D + k], __float_as_uint(v));
}

__global__ void scatter_mean_pre(const float* __restrict__ hm,
                                 const int* __restrict__ batch,
                                 float* __restrict__ gsum,
                                 float* __restrict__ cnt, int n) {
    long idx = (long)blockIdx.x * blockDim.x + threadIdx.x;
    if (idx >= (long)n * 32) return;
    int c = (int)(idx / 32);
    int k = (int)(idx % 32);
    int b = batch[c];
    atomicAdd(&gsum[b * 32 + k], hm[idx]);
    if (k == 0) atomicAdd(&cnt[b], 1.0f);
}

// MLP head: out[b] = relu(g[b,:] @ Wfc1 + bfc1) @ Wfc2 + bfc2, g = gsum/max(cnt,1)
__global__ __launch_bounds__(64) void head_kernel(const float* __restrict__ gsum,
                                                  const float* __restrict__ cnt,
                                                  const float* __restrict__ Wfc1,
                                                  const float* __restrict__ bfc1,
                                                  const float* __restrict__ Wfc2,
                                                  const float* __restrict__ bfc2,
                                                  float* __restrict__ outp) {
    int b = threadIdx.x;   // 0..63
    float g[32];
    float inv = 1.0f / fmaxf(cnt[b], 1.0f);
#pragma unroll
    for (int k = 0; k < 32; ++k) g[k] = gsum[b * 32 + k] * inv;
    float acc = 0.0f;
    for (int j = 0; j < 64; ++j) {
        float t = bfc1[j];
#pragma unroll
        for (int k = 0; k < 32; ++k) t += g[k] * Wfc1[k * 64 + j];
        t = (t > 0.0f) ? t : 0.0f;
        acc += t * Wfc2[j];
    }
    outp[b] = acc + bfc2[0];
}

// -------------------- launch --------------------

static inline int cdiv(long a, long b) { return (int)((a + b - 1) / b); }

extern "C" void kernel_launch(void* const* d_in, const int* in_sizes, int n_in,
                              void* d_out, int out_size, void* d_ws, size_t ws_size,
                              hipStream_t stream) {
    const float* x        = (const float*)d_in[0];
    const int*   ei       = (const int*)  d_in[1];   // [2, E]
    const float* ea       = (const float*)d_in[2];
    const int*   cl1      = (const int*)  d_in[3];
    const int*   ei2      = (const int*)  d_in[4];   // [2, E2]
    const float* ea2      = (const float*)d_in[5];
    const int*   cl2      = (const int*)  d_in[6];
    const int*   batch    = (const int*)  d_in[7];
    const float* W1       = (const float*)d_in[8];
    const float* att1     = (const float*)d_in[9];
    const float* b1       = (const float*)d_in[10];
    const float* W2       = (const float*)d_in[11];
    const float* att2     = (const float*)d_in[12];
    const float* b2       = (const float*)d_in[13];
    const float* Wfc1     = (const float*)d_in[14];
    const float* bfc1     = (const float*)d_in[15];
    const float* Wfc2     = (const float*)d_in[16];
    const float* bfc2     = (const float*)d_in[17];
    float*       out      = (float*)d_out;

    // workspace layout (floats); conv2 reuses conv1 regions.
    // NOTE: gemm_hp_w2 relies on this layout for safe unguarded padding:
    //   - padded hp reads (rows 25000..25007) land in the adjacent hm region
    //   - padded h2 writes (<= 800,256 floats) stay inside the 1.6M-float h region
    float* ws     = (float*)d_ws;
    float* h      = ws;                         // max(N*16, C1*32) = 1,600,000
    float* logits = h + 1600000;                // E = 3,200,000 (reused for E2)
    float* mbuf   = logits + Ee;                // N (reused as m2 over C1)
    float* sbuf   = mbuf + Nn;                  // N (reused as s2 over C1)
    float* outbuf = sbuf + Nn;                  // max(N*16, C1*32) = 1,600,000
    float* hp     = outbuf + 1600000;           // C1*16 = 400,000
    float* hm     = hp + (long)C1n * 16;        // C2*32 = 200,000
    float* gsum   = hm + (long)C2n * 32;        // B*32
    float* cnt    = gsum + (long)Bn * 32;       // B

    const int T = 256;
    const float NEG_INF = -INFINITY;

    // ---- conv1 ----
    fill_kernel<<<cdiv(Nn, T), T, 0, stream>>>(mbuf, NEG_INF, Nn);
    fill_kernel<<<cdiv(Nn, T), T, 0, stream>>>(sbuf, 0.0f, Nn);
    fill_kernel<<<cdiv((long)Nn * 16, T), T, 0, stream>>>(outbuf, 0.0f, (long)Nn * 16);
    fill_kernel<<<cdiv((long)C1n * 16, T), T, 0, stream>>>(hp, 0.0f, (long)C1n * 16);

    gemm_x_w1<<<Nn / 16, 32, 0, stream>>>(x, W1, h);

    edge_logits_max<16><<<cdiv(Ee, T), T, 0, stream>>>(h, ei, att1, logits, mbuf, Ee);
    edge_exp_sum<<<cdiv(Ee, T), T, 0, stream>>>(ei, logits, mbuf, sbuf, Ee);
    edge_aggregate<16><<<cdiv(Ee, T), T, 0, stream>>>(h, ei, ea, logits, sbuf, outbuf, Ee);

    bias_relu_pool<16><<<cdiv((long)Nn * 16, T), T, 0, stream>>>(outbuf, b1, cl1, hp, Nn);

    // ---- conv2 (reuse m/s/out/logits/h regions) ----
    fill_kernel<<<cdiv(C1n, T), T, 0, stream>>>(mbuf, NEG_INF, C1n);
    fill_kernel<<<cdiv(C1n, T), T, 0, stream>>>(sbuf, 0.0f, C1n);
    fill_kernel<<<cdiv((long)C1n * 32, T), T, 0, stream>>>(outbuf, 0.0f, (long)C1n * 32);
    fill_kernel<<<cdiv((long)C2n * 32, T), T, 0, stream>>>(hm, 0.0f, (long)C2n * 32);
    fill_kernel<<<cdiv((long)Bn * 32 + Bn, T), T, 0, stream>>>(gsum, 0.0f, (long)Bn * 32 + Bn);

    gemm_hp_w2<<<cdiv(C1n, 16), 32, 0, stream>>>(hp, W2, h);

    edge_logits_max<32><<<cdiv(E2n, T), T, 0, stream>>>(h, ei2, att2, logits, mbuf, E2n);
    edge_exp_sum<<<cdiv(E2n, T), T, 0, stream>>>(ei2, logits, mbuf, sbuf, E2n);
    edge_aggregate<32><<<cdiv(E2n, T), T, 0, stream>>>(h, ei2, ea2, logits, sbuf, outbuf, E2n);

    bias_relu_pool<32><<<cdiv((long)C1n * 32, T), T, 0, stream>>>(outbuf, b2, cl2, hm, C1n);

    scatter_mean_pre<<<cdiv((long)C2n * 32, T), T, 0, stream>>>(hm, batch, gsum, cnt, C2n);

    head_kernel<<<1, 64, 0, stream>>>(gsum, cnt, Wfc1, bfc1, Wfc2, bfc2, out);
}